// DNC_2542620639283
// MI455X (gfx1250) — compile-verified
//
#include <hip/hip_runtime.h>
#include <math.h>

// ---------------- problem constants ----------------
#define Bsz 256
#define NN  2048
#define Dd  128
#define Rr  4
#define Hh  1024
#define Ii  1024
#define CIs 1536              // I + R*D
#define IFs 787               // interface size
#define OPIs 1536             // H + R*D

typedef __attribute__((ext_vector_type(2))) float v2f;
typedef __attribute__((ext_vector_type(8))) float v8f;

__device__ __forceinline__ float sigm(float x) { return 1.0f / (1.0f + __expf(-x)); }

// ---------------------------------------------------------------------------
// fp32 WMMA GEMM, 4 M-tiles (64 rows) x 1 N-tile (16 cols) per wave,
// software-pipelined: next k-step fragments are loaded before the current
// step's 4 V_WMMA_F32_16X16X4_F32 issue, so loads overlap matrix ops.
//   C[m,n] = scale * sum_k Aelem(m,k)*Belem(k,n) + bias[n]
//   AT=false: Aelem = A[m*lda+k]      AT=true: Aelem = A[k*lda+m]
//   BT=true : Belem = B[n*ldb+k]      BT=false: Belem = B[k*ldb+n]
// f32 A 16x4 layout: lanes 0-15 -> M=lane, {v0:K0, v1:K1}; lanes 16-31 -> {K2,K3}
// f32 B 4x16 layout: lanes 0-15 -> N=lane, {v0:K0, v1:K1}; lanes 16-31 -> {K2,K3}
// C/D: VGPR v -> M = v (lanes 0-15) / v+8 (lanes 16-31), N = lane&15
// ---------------------------------------------------------------------------
template <bool AT, bool BT>
__global__ __launch_bounds__(32) void wmma_gemm_f32(
    const float* __restrict__ A, const float* __restrict__ Bm,
    const float* __restrict__ bias, float* __restrict__ C,
    int M, int N, int K, int lda, int ldb, int ldc, float scale)
{
  const int lane = threadIdx.x & 31;
  const int half = lane >> 4;
  const int l16  = lane & 15;
  const int tm = blockIdx.y * 64;
  const int tn = blockIdx.x * 16;

  int am0 = tm + l16;      if (am0 > M - 1) am0 = M - 1;   // clamps; stores guarded
  int am1 = tm + 16 + l16; if (am1 > M - 1) am1 = M - 1;
  int am2 = tm + 32 + l16; if (am2 > M - 1) am2 = M - 1;
  int am3 = tm + 48 + l16; if (am3 > M - 1) am3 = M - 1;
  int bn  = tn + l16;      if (bn  > N - 1) bn  = N - 1;

  // per-lane base pointers (hoisted 64-bit address math)
  const float* pa0 = AT ? (A + am0) : (A + am0 * lda);
  const float* pa1 = AT ? (A + am1) : (A + am1 * lda);
  const float* pa2 = AT ? (A + am2) : (A + am2 * lda);
  const float* pa3 = AT ? (A + am3) : (A + am3 * lda);
  const float* pb  = BT ? (Bm + bn * ldb) : (Bm + bn);

  auto ldA = [&](const float* p, int ka) {
    v2f a;
    if (AT) { a.x = p[ka * lda]; a.y = p[(ka + 1) * lda]; }
    else    { a.x = p[ka];       a.y = p[ka + 1]; }
    return a;
  };
  auto ldB = [&](int ka) {
    v2f b;
    if (BT) { b.x = pb[ka];       b.y = pb[ka + 1]; }
    else    { b.x = pb[ka * ldb]; b.y = pb[(ka + 1) * ldb]; }
    return b;
  };

  v8f acc0 = {}, acc1 = {}, acc2 = {}, acc3 = {};

  // prologue: fragments for k-step 0
  int ka = 2 * half;
  v2f bc  = ldB(ka);
  v2f a0c = ldA(pa0, ka), a1c = ldA(pa1, ka), a2c = ldA(pa2, ka), a3c = ldA(pa3, ka);

  for (int k0 = 4; k0 < K; k0 += 4) {
    const int kn = k0 + 2 * half;
    // issue next-step loads before consuming current fragments
    v2f bnx  = ldB(kn);
    v2f a0n = ldA(pa0, kn), a1n = ldA(pa1, kn), a2n = ldA(pa2, kn), a3n = ldA(pa3, kn);
    acc0 = __builtin_amdgcn_wmma_f32_16x16x4_f32(false, a0c, false, bc, (short)0, acc0, false, false);
    acc1 = __builtin_amdgcn_wmma_f32_16x16x4_f32(false, a1c, false, bc, (short)0, acc1, false, false);
    acc2 = __builtin_amdgcn_wmma_f32_16x16x4_f32(false, a2c, false, bc, (short)0, acc2, false, false);
    acc3 = __builtin_amdgcn_wmma_f32_16x16x4_f32(false, a3c, false, bc, (short)0, acc3, false, false);
    bc = bnx; a0c = a0n; a1c = a1n; a2c = a2n; a3c = a3n;
  }
  // epilogue
  acc0 = __builtin_amdgcn_wmma_f32_16x16x4_f32(false, a0c, false, bc, (short)0, acc0, false, false);
  acc1 = __builtin_amdgcn_wmma_f32_16x16x4_f32(false, a1c, false, bc, (short)0, acc1, false, false);
  acc2 = __builtin_amdgcn_wmma_f32_16x16x4_f32(false, a2c, false, bc, (short)0, acc2, false, false);
  acc3 = __builtin_amdgcn_wmma_f32_16x16x4_f32(false, a3c, false, bc, (short)0, acc3, false, false);

  const int scol = tn + l16;
  if (scol < N) {
    const float bv = bias ? bias[scol] : 0.0f;
    v8f accs[4] = {acc0, acc1, acc2, acc3};
#pragma unroll
    for (int t = 0; t < 4; ++t) {
      const int rbase = tm + t * 16 + half * 8;
#pragma unroll
      for (int v = 0; v < 8; ++v) {
        const int r = rbase + v;
        if (r < M) C[r * ldc + scol] = accs[t][v] * scale + bv;
      }
    }
  }
}

// ---------------- small kernels ----------------

// column mean of memory (N x D) -> colmean[D]
__global__ void k_colmean(const float* __restrict__ mem, float* __restrict__ cm) {
  int d = threadIdx.x;  // 128
  float s = 0.0f;
  for (int n = 0; n < NN; ++n) s += mem[n * Dd + d];
  cm[d] = s * (1.0f / NN);
}

// cin = [x | broadcast colmean x R]
__global__ void k_build_cin(const float* __restrict__ x, const float* __restrict__ cm,
                            float* __restrict__ cin) {
  int i = blockIdx.x * blockDim.x + threadIdx.x;
  if (i >= Bsz * CIs) return;
  int b = i / CIs, k = i % CIs;
  cin[i] = (k < Ii) ? x[b * Ii + k] : cm[(k - Ii) & (Dd - 1)];
}

__global__ void k_bias2(const float* __restrict__ a, const float* __restrict__ b,
                        float* __restrict__ o, int n) {
  int i = blockIdx.x * blockDim.x + threadIdx.x;
  if (i < n) o[i] = a[i] + b[i];
}

// LSTM gates (i,f,g,o order; c0=h0=0) -> h, and h into cat[:, :H]
__global__ void k_gates(const float* __restrict__ g, float* __restrict__ h,
                        float* __restrict__ cat) {
  int i = blockIdx.x * blockDim.x + threadIdx.x;
  if (i >= Bsz * Hh) return;
  int b = i / Hh, j = i % Hh;
  const float* gr = g + b * 4 * Hh;
  float c  = sigm(gr[j]) * tanhf(gr[2 * Hh + j]);
  float hv = sigm(gr[3 * Hh + j]) * tanhf(c);
  h[i] = hv;
  cat[b * OPIs + j] = hv;
}

// parse interface vector; layout offsets for W=1,R=4,D=128
__global__ void k_parse(const float* __restrict__ itf, float* __restrict__ wv,
                        float* __restrict__ er, float* __restrict__ wg,
                        float* __restrict__ ag, float* __restrict__ rmodes,
                        float* __restrict__ rstr, float* __restrict__ rkeys,
                        float* __restrict__ evav) {
  int b = blockIdx.x, d = threadIdx.x;  // 128 threads
  const float* row = itf + b * IFs;
  __shared__ float s_wg;
  float wvv = row[d];
  float erv = sigm(row[Dd + d]);
  wv[b * Dd + d] = wvv;
  er[b * Dd + d] = erv;
  if (d == 0) {
    float w = sigm(row[256]);
    wg[b] = w; s_wg = w;
    ag[b] = sigm(row[257]);           // row[258] = free gate, unused
  }
  if (d < Rr) {
    float m0 = row[259 + d * 3], m1 = row[260 + d * 3], m2 = row[261 + d * 3];
    float mx = fmaxf(m0, fmaxf(m1, m2));
    float e0 = __expf(m0 - mx), e1 = __expf(m1 - mx), e2 = __expf(m2 - mx);
    float s = e0 + e1 + e2;
    rmodes[(b * Rr + d) * 3 + 0] = e0 / s;
    rmodes[(b * Rr + d) * 3 + 1] = e1 / s;
    rmodes[(b * Rr + d) * 3 + 2] = e2 / s;
    float rs = row[271 + d];
    rstr[b * Rr + d] = (rs > 20.0f) ? rs : log1pf(__expf(rs));   // softplus
  }
  for (int r = 0; r < Rr; ++r)
    rkeys[(b * Rr + r) * Dd + d] = row[275 + r * Dd + d];
  __syncthreads();
  float w = s_wg;
  evav[b * 2 * Dd + d]      = erv * w;   // erase * gate
  evav[b * 2 * Dd + Dd + d] = wvv * w;   // write_val * gate
}

// L2-normalize rows of (rows x 128)
__global__ void k_normrows(const float* __restrict__ in, float* __restrict__ out) {
  int r = blockIdx.x, d = threadIdx.x;  // 128 threads
  __shared__ float red[128];
  float v = in[r * Dd + d];
  red[d] = v * v; __syncthreads();
  for (int s = 64; s > 0; s >>= 1) { if (d < s) red[d] += red[d + s]; __syncthreads(); }
  float nrm = fmaxf(sqrtf(red[0]), 1e-12f);
  out[r * Dd + d] = v / nrm;
}

// allocation weighting via rank (stable argsort) + cumprod, single block
__global__ void k_alloc(const float* __restrict__ usage, float* __restrict__ alloc) {
  __shared__ float us[NN];
  __shared__ float su[NN];
  __shared__ float cp[NN];
  __shared__ int   rk[NN];
  int t = threadIdx.x;  // 256
  for (int i = t; i < NN; i += 256) us[i] = usage[i];
  __syncthreads();
  for (int i = t; i < NN; i += 256) {
    float ui = us[i];
    int rank = 0;
    for (int j = 0; j < NN; ++j) {
      float uj = us[j];
      rank += (uj < ui) || (uj == ui && j < i);
    }
    rk[i] = rank;
    su[rank] = ui;
  }
  __syncthreads();
  if (t == 0) {
    float p = 1.0f;
    for (int k = 0; k < NN; ++k) { p *= (1.0f - su[k]); cp[k] = p; }
  }
  __syncthreads();
  for (int i = t; i < NN; i += 256) {
    int r = rk[i];
    alloc[i] = (r == 0) ? su[0] : su[r] * cp[r - 1];
  }
}

// row softmax in place, 256 threads/row
__global__ void k_softmax(float* __restrict__ x, int cols) {
  int r = blockIdx.x, t = threadIdx.x;
  float* row = x + r * cols;
  __shared__ float red[256];
  float mx = -3.4e38f;
  for (int c = t; c < cols; c += 256) mx = fmaxf(mx, row[c]);
  red[t] = mx; __syncthreads();
  for (int s = 128; s > 0; s >>= 1) { if (t < s) red[t] = fmaxf(red[t], red[t + s]); __syncthreads(); }
  mx = red[0]; __syncthreads();
  float sm = 0.0f;
  for (int c = t; c < cols; c += 256) { float e = __expf(row[c] - mx); row[c] = e; sm += e; }
  red[t] = sm; __syncthreads();
  for (int s = 128; s > 0; s >>= 1) { if (t < s) red[t] += red[t + s]; __syncthreads(); }
  float inv = 1.0f / red[0];
  for (int c = t; c < cols; c += 256) row[c] *= inv;
}

// write_w[b,n] = wg[b]*(0.5*cw[b,n] + 0.5*alloc[n]*ag[b])
__global__ void k_write_w(const float* __restrict__ cw, const float* __restrict__ alloc,
                          const float* __restrict__ wg, const float* __restrict__ ag,
                          float* __restrict__ w) {
  int i = blockIdx.x * blockDim.x + threadIdx.x;
  if (i >= Bsz * NN) return;
  int b = i / NN, n = i % NN;
  w[i] = wg[b] * (0.5f * cw[i] + 0.5f * alloc[n] * ag[b]);
}

// s[b] = sum_n w[b,n]
__global__ void k_rowsum(const float* __restrict__ w, float* __restrict__ s) {
  int b = blockIdx.x, t = threadIdx.x;
  __shared__ float red[256];
  float acc = 0.0f;
  for (int n = t; n < NN; n += 256) acc += w[b * NN + n];
  red[t] = acc; __syncthreads();
  for (int st = 128; st > 0; st >>= 1) { if (t < st) red[t] += red[t + st]; __syncthreads(); }
  if (t == 0) s[b] = red[0];
}

// lu row/col mean (symmetric, diag removed): (1/(N*B)) sum_b w[b,n]*(s[b]-w[b,n])
__global__ void k_lu_mean(const float* __restrict__ w, const float* __restrict__ s,
                          float* __restrict__ lum) {
  __shared__ float ss[Bsz];
  int t = threadIdx.x;  // 128
  for (int b = t; b < Bsz; b += blockDim.x) ss[b] = s[b];
  __syncthreads();
  int n = blockIdx.x * blockDim.x + t;
  float acc = 0.0f;
  for (int b = 0; b < Bsz; ++b) {
    float wv = w[b * NN + n];
    acc += wv * (ss[b] - wv);
  }
  lum[n] = acc * (1.0f / ((float)NN * (float)Bsz));
}

// column mean of link (coalesced across threads)
__global__ void k_link_colmean(const float* __restrict__ link, float* __restrict__ cm) {
  int m = blockIdx.x * blockDim.x + threadIdx.x;
  if (m >= NN) return;
  float acc = 0.0f;
  for (int n = 0; n < NN; ++n) acc += link[n * NN + m];
  cm[m] = acc * (1.0f / NN);
}

// row mean of link (one block per row)
__global__ void k_link_rowmean(const float* __restrict__ link, float* __restrict__ rm) {
  int n = blockIdx.x, t = threadIdx.x;
  __shared__ float red[256];
  float acc = 0.0f;
  for (int m = t; m < NN; m += 256) acc += link[n * NN + m];
  red[t] = acc; __syncthreads();
  for (int s = 128; s > 0; s >>= 1) { if (t < s) red[t] += red[t + s]; __syncthreads(); }
  if (t == 0) rm[n] = red[0] * (1.0f / NN);
}

// bwd = 0.9*colmean(link)+0.1*lu_mean ; fwd = 0.9*rowmean(link)+0.1*lu_mean
__global__ void k_bwdfwd(const float* __restrict__ cm, const float* __restrict__ rm,
                         const float* __restrict__ lum, float* __restrict__ bwd,
                         float* __restrict__ fwd) {
  int i = blockIdx.x * blockDim.x + threadIdx.x;
  if (i >= NN) return;
  bwd[i] = 0.9f * cm[i] + 0.1f * lum[i];
  fwd[i] = 0.9f * rm[i] + 0.1f * lum[i];
}

// mem_new = memory*(1 - erase_mat) + add_mat   (evavmat is N x 2D: [erase|add])
__global__ void k_memnew(const float* __restrict__ mem, const float* __restrict__ evavmat,
                         float* __restrict__ memnew) {
  int i = blockIdx.x * blockDim.x + threadIdx.x;
  if (i >= NN * Dd) return;
  int n = i / Dd, d = i % Dd;
  float e = evavmat[n * 2 * Dd + d];
  float a = evavmat[n * 2 * Dd + Dd + d];
  memnew[i] = mem[i] * (1.0f - e) + a;
}

// softmax(sim*str) then combine with modes and bwd/fwd, in place -> new_rw
__global__ void k_softmax_combine(float* __restrict__ sim, const float* __restrict__ rstr,
                                  const float* __restrict__ rmodes,
                                  const float* __restrict__ bwd,
                                  const float* __restrict__ fwd) {
  int m = blockIdx.x, t = threadIdx.x;  // 1024 rows, 256 threads
  float str = rstr[m];
  float r0 = rmodes[m * 3 + 0], r1 = rmodes[m * 3 + 1], r2 = rmodes[m * 3 + 2];
  float* row = sim + m * NN;
  __shared__ float red[256];
  float mx = -3.4e38f;
  for (int c = t; c < NN; c += 256) mx = fmaxf(mx, row[c] * str);
  red[t] = mx; __syncthreads();
  for (int s = 128; s > 0; s >>= 1) { if (t < s) red[t] = fmaxf(red[t], red[t + s]); __syncthreads(); }
  mx = red[0]; __syncthreads();
  float sm = 0.0f;
  for (int c = t; c < NN; c += 256) sm += __expf(row[c] * str - mx);
  red[t] = sm; __syncthreads();
  for (int s = 128; s > 0; s >>= 1) { if (t < s) red[t] += red[t + s]; __syncthreads(); }
  float inv = 1.0f / red[0];
  for (int c = t; c < NN; c += 256) {
    float content = __expf(row[c] * str - mx) * inv;
    row[c] = r0 * content + r1 * bwd[c] + r2 * fwd[c];
  }
}

// read_out (B*R, D) -> cat[:, H:H+R*D]
__global__ void k_copy_ro(const float* __restrict__ ro, float* __restrict__ cat) {
  int i = blockIdx.x * blockDim.x + threadIdx.x;
  if (i >= Bsz * Rr * Dd) return;
  int b = i / (Rr * Dd), j = i % (Rr * Dd);
  cat[b * OPIs + Hh + j] = ro[i];
}

// ---------------- host side ----------------
static inline dim3 tiles64(int N, int M) { return dim3((N + 15) / 16, (M + 63) / 64); }
static inline int cdiv(int a, int b) { return (a + b - 1) / b; }

extern "C" void kernel_launch(void* const* d_in, const int* in_sizes, int n_in,
                              void* d_out, int out_size, void* d_ws, size_t ws_size,
                              hipStream_t stream) {
  const float* x      = (const float*)d_in[0];
  const float* memory = (const float*)d_in[1];
  const float* usage  = (const float*)d_in[2];
  const float* link   = (const float*)d_in[3];
  const float* W_ih   = (const float*)d_in[4];
  // d_in[5] = W_hh (unused: h0 = 0)
  const float* b_ih   = (const float*)d_in[6];
  const float* b_hh   = (const float*)d_in[7];
  const float* W_if   = (const float*)d_in[8];
  const float* b_if   = (const float*)d_in[9];
  const float* W_out  = (const float*)d_in[10];
  const float* b_out  = (const float*)d_in[11];
  float* out = (float*)d_out;

  // bump allocator over d_ws
  char* wp = (char*)d_ws;
  auto alloc_f = [&](size_t nfloats) {
    float* p = (float*)wp;
    wp += ((nfloats * sizeof(float) + 255) / 256) * 256;
    return p;
  };
  float* colmean = alloc_f(Dd);
  float* cin     = alloc_f((size_t)Bsz * CIs);
  float* bsum    = alloc_f(4 * Hh);
  float* g       = alloc_f((size_t)Bsz * 4 * Hh);
  float* h       = alloc_f((size_t)Bsz * Hh);
  float* itf     = alloc_f((size_t)Bsz * IFs);
  float* wv      = alloc_f((size_t)Bsz * Dd);
  float* er      = alloc_f((size_t)Bsz * Dd);
  float* wg      = alloc_f(Bsz);
  float* ag      = alloc_f(Bsz);
  float* rmodes  = alloc_f((size_t)Bsz * Rr * 3);
  float* rstr    = alloc_f((size_t)Bsz * Rr);
  float* rkeys   = alloc_f((size_t)Bsz * Rr * Dd);
  float* evav    = alloc_f((size_t)Bsz * 2 * Dd);
  float* wkeyn   = alloc_f((size_t)Bsz * Dd);
  float* memn    = alloc_f((size_t)NN * Dd);
  float* cw      = alloc_f((size_t)Bsz * NN);
  float* allocv  = alloc_f(NN);
  float* writew  = alloc_f((size_t)Bsz * NN);
  float* srow    = alloc_f(Bsz);
  float* lumean  = alloc_f(NN);
  float* colml   = alloc_f(NN);
  float* rowml   = alloc_f(NN);
  float* bwd     = alloc_f(NN);
  float* fwd     = alloc_f(NN);
  float* evavmat = alloc_f((size_t)NN * 2 * Dd);
  float* memnew  = alloc_f((size_t)NN * Dd);
  float* memnewn = alloc_f((size_t)NN * Dd);
  float* rkn     = alloc_f((size_t)Bsz * Rr * Dd);
  float* sim     = alloc_f((size_t)Bsz * Rr * NN);   // becomes new_rw in place
  float* ro      = alloc_f((size_t)Bsz * Rr * Dd);
  float* cat     = alloc_f((size_t)Bsz * OPIs);
  (void)ws_size; (void)in_sizes; (void)n_in; (void)out_size;

  // 1) read_flat via column mean (prev_rw uniform), build controller input
  k_colmean<<<1, 128, 0, stream>>>(memory, colmean);
  k_build_cin<<<cdiv(Bsz * CIs, 256), 256, 0, stream>>>(x, colmean, cin);
  k_bias2<<<cdiv(4 * Hh, 256), 256, 0, stream>>>(b_ih, b_hh, bsum, 4 * Hh);

  // 2) controller GEMM + gates
  wmma_gemm_f32<false, true><<<tiles64(4 * Hh, Bsz), 32, 0, stream>>>(
      cin, W_ih, bsum, g, Bsz, 4 * Hh, CIs, CIs, CIs, 4 * Hh, 1.0f);
  k_gates<<<cdiv(Bsz * Hh, 256), 256, 0, stream>>>(g, h, cat);

  // 3) interface projection + parse
  wmma_gemm_f32<false, true><<<tiles64(IFs, Bsz), 32, 0, stream>>>(
      h, W_if, b_if, itf, Bsz, IFs, Hh, Hh, Hh, IFs, 1.0f);
  k_parse<<<Bsz, 128, 0, stream>>>(itf, wv, er, wg, ag, rmodes, rstr, rkeys, evav);

  // 4) write content addressing
  k_normrows<<<Bsz, 128, 0, stream>>>(wv, wkeyn);
  k_normrows<<<NN, 128, 0, stream>>>(memory, memn);
  wmma_gemm_f32<false, true><<<tiles64(NN, Bsz), 32, 0, stream>>>(
      wkeyn, memn, nullptr, cw, Bsz, NN, Dd, Dd, Dd, NN, 1.0f);
  k_softmax<<<Bsz, 256, 0, stream>>>(cw, NN);

  // 5) allocation + write weights
  k_alloc<<<1, 256, 0, stream>>>(usage, allocv);
  k_write_w<<<cdiv(Bsz * NN, 256), 256, 0, stream>>>(cw, allocv, wg, ag, writew);

  // 6) link means (no N x N materialization: lu is symmetric, prev_rw uniform)
  k_rowsum<<<Bsz, 256, 0, stream>>>(writew, srow);
  k_lu_mean<<<NN / 128, 128, 0, stream>>>(writew, srow, lumean);
  k_link_colmean<<<NN / 256, 256, 0, stream>>>(link, colml);
  k_link_rowmean<<<NN, 256, 0, stream>>>(link, rowml);
  k_bwdfwd<<<cdiv(NN, 256), 256, 0, stream>>>(colml, rowml, lumean, bwd, fwd);

  // 7) memory write: [erase_mat | add_mat] = write_w^T @ [ev|av] / B
  wmma_gemm_f32<true, false><<<tiles64(2 * Dd, NN), 32, 0, stream>>>(
      writew, evav, nullptr, evavmat, NN, 2 * Dd, Bsz, NN, 2 * Dd, 2 * Dd,
      1.0f / (float)Bsz);
  k_memnew<<<cdiv(NN * Dd, 256), 256, 0, stream>>>(memory, evavmat, memnew);

  // 8) read addressing on updated memory
  k_normrows<<<NN, 128, 0, stream>>>(memnew, memnewn);
  k_normrows<<<Bsz * Rr, 128, 0, stream>>>(rkeys, rkn);
  wmma_gemm_f32<false, true><<<tiles64(NN, Bsz * Rr), 32, 0, stream>>>(
      rkn, memnewn, nullptr, sim, Bsz * Rr, NN, Dd, Dd, Dd, NN, 1.0f);
  k_softmax_combine<<<Bsz * Rr, 256, 0, stream>>>(sim, rstr, rmodes, bwd, fwd);

  // 9) read vectors + output projection
  wmma_gemm_f32<false, false><<<tiles64(Dd, Bsz * Rr), 32, 0, stream>>>(
      sim, memnew, nullptr, ro, Bsz * Rr, Dd, NN, NN, Dd, Dd, 1.0f);
  k_copy_ro<<<cdiv(Bsz * Rr * Dd, 256), 256, 0, stream>>>(ro, cat);
  wmma_gemm_f32<false, true><<<tiles64(Hh, Bsz), 32, 0, stream>>>(
      cat, W_out, b_out, out, Bsz, Hh, OPIs, OPIs, OPIs, Hh, 1.0f);
}